// GINVirtual_node_19602230739663
// MI455X (gfx1250) — compile-verified
//
#include <hip/hip_runtime.h>
#include <hip/hip_bf16.h>

typedef float v2f __attribute__((ext_vector_type(2)));
typedef float v8f __attribute__((ext_vector_type(8)));

#define N_NODES 100000
#define N_EDGES 200000
#define N_GRAPHS 2500
#define NGP 2512          // N_GRAPHS padded to multiple of 16
#define EMB 300
#define EMBP 304          // padded to multiple of 16
#define HID 600
#define HIDP 608          // padded to multiple of 16
#define KATOM 176         // 173 real rows padded to multiple of 16
#define BN_EPS 1e-5f

static __device__ __forceinline__ void bond_softmax(float a, int d, float* p) {
    float s = a / (float)(d - 1);
    float m = fmaxf(s * (float)(d - 1), 0.0f);
    float sum = 0.0f;
    for (int j = 0; j < d; ++j) { float e = __expf(s * (float)j - m); p[j] = e; sum += e; }
    float inv = 1.0f / sum;
    for (int j = 0; j < d; ++j) p[j] *= inv;
}

// ---------------- weight staging ----------------

// WcatT[c*KATOM + k] : transposed+padded concat of the 9 atom tables, c<EMBP, k<KATOM
__global__ void build_wcatT_kernel(const float* w0, const float* w1, const float* w2,
                                   const float* w3, const float* w4, const float* w5,
                                   const float* w6, const float* w7, const float* w8,
                                   float* __restrict__ W) {
    int idx = blockIdx.x * blockDim.x + threadIdx.x;
    if (idx >= EMBP * KATOM) return;
    int c = idx / KATOM, k = idx - c * KATOM;
    const int dims[9] = {119, 4, 12, 12, 10, 6, 6, 2, 2};
    const float* ws[9] = {w0, w1, w2, w3, w4, w5, w6, w7, w8};
    float v = 0.0f;
    if (c < EMB && k < 173) {
        int off = 0;
        for (int i = 0; i < 9; ++i) {
            if (k >= off && k < off + dims[i]) v = ws[i][(k - off) * EMB + c];
            off += dims[i];
        }
    }
    W[idx] = v;
}

// Wt[c*KP + k] = (k<K && c<cols) ? W[k*cols + c] : 0
__global__ void build_wt_kernel(const float* __restrict__ W, float* __restrict__ Wt,
                                int K, int cols, int KP, int colsP) {
    int idx = blockIdx.x * blockDim.x + threadIdx.x;
    if (idx >= colsP * KP) return;
    int c = idx / KP, k = idx - c * KP;
    Wt[idx] = (k < K && c < cols) ? W[k * cols + c] : 0.0f;
}

__global__ void pcat_kernel(const float* __restrict__ x, float* __restrict__ P) {
    int n = blockIdx.x * blockDim.x + threadIdx.x;
    if (n >= N_NODES) return;
    const int dims[9] = {119, 4, 12, 12, 10, 6, 6, 2, 2};
    int off = 0;
    float* row = P + (size_t)n * KATOM;
    for (int i = 0; i < 9; ++i) {
        int d = dims[i];
        float a = x[n * 9 + i];
        float s = a / (float)(d - 1);
        float m = fmaxf(s * (float)(d - 1), 0.0f);
        float sum = 0.0f;
        for (int j = 0; j < d; ++j) sum += __expf(s * (float)j - m);
        float inv = 1.0f / sum;
        for (int j = 0; j < d; ++j) row[off + j] = __expf(s * (float)j - m) * inv;
        off += d;
    }
    row[173] = 0.0f; row[174] = 0.0f; row[175] = 0.0f;
}

// ---------------- f32 WMMA GEMM ----------------
// C[M,colsP] = A[M,KP] @ Bt^T (+bias), Bt[colsP,KP] is the transposed padded weight.
// M, KP, colsP multiples of 16 -> zero predication in the K loop.
// One wave computes one 16x16 C tile with V_WMMA_F32_16X16X4_F32.

__global__ void gemm_wmma_f32(const float* __restrict__ A, const float* __restrict__ Bt,
                              const float* __restrict__ bias, float* __restrict__ C,
                              int M, int KP, int colsP, int realcols) {
    int wid = (blockIdx.x * blockDim.x + threadIdx.x) >> 5;   // wave-uniform
    int lane = threadIdx.x & 31;
    int tilesM = M >> 4;
    int tilesN = colsP >> 4;
    if (wid >= tilesM * tilesN) return;                        // whole-wave exit
    int m0 = (wid % tilesM) << 4;
    int n0 = (wid / tilesM) << 4;
    int half = lane >> 4;          // lanes 0-15: K pair {0,1}; lanes 16-31: {2,3}
    int lr   = lane & 15;
    const v2f* __restrict__ ap = (const v2f*)(A  + (size_t)(m0 + lr) * KP + 2 * half);
    const v2f* __restrict__ bp = (const v2f*)(Bt + (size_t)(n0 + lr) * KP + 2 * half);
    v8f acc = {};
#pragma unroll 4
    for (int k = 0; k < KP; k += 4) {
        v2f a = ap[0];
        v2f b = bp[0];
        acc = __builtin_amdgcn_wmma_f32_16x16x4_f32(false, a, false, b,
                                                    (short)0, acc, false, false);
        ap += 2;   // advance 4 floats
        bp += 2;
    }
    int bcol = n0 + lr;
    float bv = (bias != nullptr && bcol < realcols) ? bias[bcol] : 0.0f;
    bool pad = bcol >= realcols;
#pragma unroll
    for (int i = 0; i < 8; ++i) {
        int row = m0 + i + 8 * half;   // VGPR i: lanes0-15 -> M=i, lanes16-31 -> M=i+8
        C[(size_t)row * colsP + bcol] = pad ? 0.0f : (acc[i] + bv);
    }
}

// ---------------- per-layer elementwise / scatter kernels ----------------

__global__ void init_vn_kernel(const float* __restrict__ vn_w, float* __restrict__ vn) {
    int idx = blockIdx.x * blockDim.x + threadIdx.x;
    if (idx >= NGP * EMBP) return;
    int g = idx / EMBP, c = idx - g * EMBP;
    vn[idx] = (g < N_GRAPHS && c < EMB) ? vn_w[c] : 0.0f;
}

__global__ void addvn_kernel(float* __restrict__ h, const float* __restrict__ vn,
                             const int* __restrict__ batch, const float* __restrict__ eps,
                             int l, float* __restrict__ agg) {
    int idx = blockIdx.x * blockDim.x + threadIdx.x;
    if (idx >= N_NODES * EMBP) return;
    int n = idx / EMBP, c = idx - n * EMBP;
    if (c >= EMB) { h[idx] = 0.0f; agg[idx] = 0.0f; return; }
    float v = h[idx] + vn[batch[n] * EMBP + c];
    h[idx] = v;
    agg[idx] = (1.0f + eps[l]) * v;
}

// wave-per-edge: bond soft-onehot fused with message + scatter-add
__global__ void edge_msg_kernel(const float* __restrict__ h, const int* __restrict__ ei,
                                const float* __restrict__ ea,
                                const float* __restrict__ w0,   // [5,300]
                                const float* __restrict__ w1,   // [6,300]
                                const float* __restrict__ w2,   // [2,300]
                                float* __restrict__ agg) {
    int wid = (blockIdx.x * blockDim.x + threadIdx.x) >> 5;
    int lane = threadIdx.x & 31;
    if (wid >= N_EDGES) return;
    int src = ei[wid];
    int dst = ei[N_EDGES + wid];
    float p0[5], p1[6], p2[2];
    bond_softmax(ea[wid * 3 + 0], 5, p0);
    bond_softmax(ea[wid * 3 + 1], 6, p1);
    bond_softmax(ea[wid * 3 + 2], 2, p2);
    const float* hs = h + (size_t)src * EMBP;
    float* ag = agg + (size_t)dst * EMBP;
    for (int c = lane; c < EMB; c += 32) {
        float a = hs[c];
#pragma unroll
        for (int j = 0; j < 5; ++j) a += p0[j] * w0[j * EMB + c];
#pragma unroll
        for (int j = 0; j < 6; ++j) a += p1[j] * w1[j * EMB + c];
#pragma unroll
        for (int j = 0; j < 2; ++j) a += p2[j] * w2[j * EMB + c];
        a = fmaxf(a, 0.0f);
        unsafeAtomicAdd(ag + c, a);               // global_atomic_add_f32
    }
}

__global__ void pool_kernel(const float* __restrict__ h, const int* __restrict__ batch,
                            float* __restrict__ vnt) {
    int idx = blockIdx.x * blockDim.x + threadIdx.x;
    if (idx >= N_NODES * EMB) return;
    int n = idx / EMB, c = idx - n * EMB;
    unsafeAtomicAdd(&vnt[batch[n] * EMBP + c], h[(size_t)n * EMBP + c]);
}

// ---------------- batchnorm (training-mode, biased var) ----------------

__global__ void stats_kernel(const float* __restrict__ X, float* __restrict__ sums,
                             float* __restrict__ sqs, int M, int colsP, int realcols) {
    int tx = threadIdx.x, ty = threadIdx.y;
    int col = blockIdx.x * 32 + tx;
    int rowsPer = (M + gridDim.y - 1) / gridDim.y;
    int r0 = blockIdx.y * rowsPer;
    int r1 = min(r0 + rowsPer, M);
    float s = 0.0f, q = 0.0f;
    if (col < realcols) {
        for (int r = r0 + ty; r < r1; r += 8) {
            float v = X[(size_t)r * colsP + col];
            s += v; q += v * v;
        }
    }
    __shared__ float ss[8][32];
    __shared__ float qq[8][32];
    ss[ty][tx] = s; qq[ty][tx] = q;
    __syncthreads();
    if (ty == 0 && col < realcols) {
#pragma unroll
        for (int i = 1; i < 8; ++i) { s += ss[i][tx]; q += qq[i][tx]; }
        unsafeAtomicAdd(&sums[col], s);
        unsafeAtomicAdd(&sqs[col], q);
    }
}

// Reads X with stride colsP; writes Y with stride outColsP.
// outColsP == colsP : zero pad columns (keeps the zero-pad invariant for GEMM inputs)
// outColsP == realcols : dense output (final d_out), pad columns skipped entirely
__global__ void bnapply_kernel(const float* __restrict__ X, float* __restrict__ Y,
                               const float* __restrict__ sums, const float* __restrict__ sqs,
                               const float* __restrict__ g, const float* __restrict__ b,
                               int M, int colsP, int realcols, int outColsP, int dorelu) {
    int idx = blockIdx.x * blockDim.x + threadIdx.x;
    if (idx >= M * colsP) return;
    int row = idx / colsP;
    int c = idx - row * colsP;
    if (c >= realcols) {
        if (outColsP == colsP) Y[(size_t)row * outColsP + c] = 0.0f;
        return;
    }
    float invM = 1.0f / (float)M;
    float mean = sums[c] * invM;
    float var = sqs[c] * invM - mean * mean;
    float y = (X[idx] - mean) * rsqrtf(var + BN_EPS) * g[c] + b[c];
    if (dorelu) y = fmaxf(y, 0.0f);
    Y[(size_t)row * outColsP + c] = y;
}

// ---------------- host orchestration ----------------

static inline int cdiv(long long a, long long b) { return (int)((a + b - 1) / b); }

static void run_gemm(const float* A, const float* Bt, const float* bias, float* C,
                     int M, int KP, int colsP, int realcols, hipStream_t s) {
    long long tiles = (long long)(M / 16) * (colsP / 16);
    gemm_wmma_f32<<<cdiv(tiles * 32, 256), 256, 0, s>>>(A, Bt, bias, C, M, KP, colsP, realcols);
}

static void run_bn(const float* X, float* Y, float* stat, const float* g, const float* b,
                   int M, int colsP, int realcols, int outColsP, int dorelu, hipStream_t s) {
    hipMemsetAsync(stat, 0, 2 * HIDP * sizeof(float), s);
    dim3 blk(32, 8);
    dim3 grd(cdiv(realcols, 32), 64);
    stats_kernel<<<grd, blk, 0, s>>>(X, stat, stat + HIDP, M, colsP, realcols);
    bnapply_kernel<<<cdiv((long long)M * colsP, 256), 256, 0, s>>>(
        X, Y, stat, stat + HIDP, g, b, M, colsP, realcols, outColsP, dorelu);
}

extern "C" void kernel_launch(void* const* d_in, const int* in_sizes, int n_in,
                              void* d_out, int out_size, void* d_ws, size_t ws_size,
                              hipStream_t stream) {
    const float* x      = (const float*)d_in[0];
    const int*   ei     = (const int*)d_in[1];
    const float* ea     = (const float*)d_in[2];
    const int*   batch  = (const int*)d_in[3];
    const float* aw[9];
    for (int i = 0; i < 9; ++i) aw[i] = (const float*)d_in[4 + i];
    const float* vn_w   = (const float*)d_in[13];
    const float* bw0    = (const float*)d_in[14];
    const float* bw1    = (const float*)d_in[15];
    const float* bw2    = (const float*)d_in[16];
    const float* eps    = (const float*)d_in[17];
    const float* cw1    = (const float*)d_in[18];
    const float* cb1    = (const float*)d_in[19];
    const float* cbn1_g = (const float*)d_in[20];
    const float* cbn1_b = (const float*)d_in[21];
    const float* cw2    = (const float*)d_in[22];
    const float* cb2    = (const float*)d_in[23];
    const float* bn_g   = (const float*)d_in[24];
    const float* bn_b   = (const float*)d_in[25];
    const float* vw1    = (const float*)d_in[26];
    const float* vb1    = (const float*)d_in[27];
    const float* vbn1_g = (const float*)d_in[28];
    const float* vbn1_b = (const float*)d_in[29];
    const float* vw2    = (const float*)d_in[30];
    const float* vb2    = (const float*)d_in[31];
    const float* vbn2_g = (const float*)d_in[32];
    const float* vbn2_b = (const float*)d_in[33];

    // workspace layout (floats)
    float* ws    = (float*)d_ws;
    float* h     = ws;                                   // N*EMBP
    float* agg   = h     + (size_t)N_NODES * EMBP;       // N*EMBP (also z)
    float* hid   = agg   + (size_t)N_NODES * EMBP;       // N*HIDP (also Pcat)
    float* WcatT = hid   + (size_t)N_NODES * HIDP;       // EMBP*KATOM
    float* Wt    = WcatT + (size_t)EMBP * KATOM;         // HIDP*HIDP staging (covers all)
    float* vn    = Wt    + (size_t)HIDP * HIDP;          // NGP*EMBP
    float* vnt   = vn    + (size_t)NGP * EMBP;           // NGP*EMBP
    float* vh    = vnt   + (size_t)NGP * EMBP;           // NGP*HIDP
    float* vz    = vh    + (size_t)NGP * HIDP;           // NGP*EMBP
    float* stat  = vz    + (size_t)NGP * EMBP;           // 2*HIDP

    float* Pcat = hid;  // reuse

    // ---- atom encoder: h = P @ Wcat (WMMA) ----
    build_wcatT_kernel<<<cdiv(EMBP * KATOM, 256), 256, 0, stream>>>(
        aw[0], aw[1], aw[2], aw[3], aw[4], aw[5], aw[6], aw[7], aw[8], WcatT);
    pcat_kernel<<<cdiv(N_NODES, 256), 256, 0, stream>>>(x, Pcat);
    run_gemm(Pcat, WcatT, nullptr, h, N_NODES, KATOM, EMBP, EMB, stream);

    init_vn_kernel<<<cdiv(NGP * EMBP, 256), 256, 0, stream>>>(vn_w, vn);

    for (int l = 0; l < 5; ++l) {
        addvn_kernel<<<cdiv((long long)N_NODES * EMBP, 256), 256, 0, stream>>>(
            h, vn, batch, eps, l, agg);
        edge_msg_kernel<<<cdiv((long long)N_EDGES * 32, 256), 256, 0, stream>>>(
            h, ei, ea, bw0 + (size_t)l * 5 * EMB, bw1 + (size_t)l * 6 * EMB,
            bw2 + (size_t)l * 2 * EMB, agg);

        // hid = relu(BN(agg @ cw1 + cb1))
        build_wt_kernel<<<cdiv(HIDP * EMBP, 256), 256, 0, stream>>>(
            cw1 + (size_t)l * EMB * HID, Wt, EMB, HID, EMBP, HIDP);
        run_gemm(agg, Wt, cb1 + (size_t)l * HID, hid, N_NODES, EMBP, HIDP, HID, stream);
        run_bn(hid, hid, stat, cbn1_g + (size_t)l * HID, cbn1_b + (size_t)l * HID,
               N_NODES, HIDP, HID, HIDP, 1, stream);

        // z = hid @ cw2 + cb2  (into agg)
        build_wt_kernel<<<cdiv(EMBP * HIDP, 256), 256, 0, stream>>>(
            cw2 + (size_t)l * HID * EMB, Wt, HID, EMB, HIDP, EMBP);
        run_gemm(hid, Wt, cb2 + (size_t)l * EMB, agg, N_NODES, HIDP, EMBP, EMB, stream);

        if (l < 4) {
            // virtual-node pool from pre-conv h (before h is overwritten)
            hipMemcpyAsync(vnt, vn, (size_t)NGP * EMBP * sizeof(float),
                           hipMemcpyDeviceToDevice, stream);
            pool_kernel<<<cdiv((long long)N_NODES * EMB, 256), 256, 0, stream>>>(h, batch, vnt);
            // h = relu(BN(z))
            run_bn(agg, h, stat, bn_g + (size_t)l * EMB, bn_b + (size_t)l * EMB,
                   N_NODES, EMBP, EMB, EMBP, 1, stream);
            // vn = relu(BN(relu(BN(vnt@vw1+vb1))@vw2+vb2))
            build_wt_kernel<<<cdiv(HIDP * EMBP, 256), 256, 0, stream>>>(
                vw1 + (size_t)l * EMB * HID, Wt, EMB, HID, EMBP, HIDP);
            run_gemm(vnt, Wt, vb1 + (size_t)l * HID, vh, NGP, EMBP, HIDP, HID, stream);
            run_bn(vh, vh, stat, vbn1_g + (size_t)l * HID, vbn1_b + (size_t)l * HID,
                   N_GRAPHS, HIDP, HID, HIDP, 1, stream);
            build_wt_kernel<<<cdiv(EMBP * HIDP, 256), 256, 0, stream>>>(
                vw2 + (size_t)l * HID * EMB, Wt, HID, EMB, HIDP, EMBP);
            run_gemm(vh, Wt, vb2 + (size_t)l * EMB, vz, NGP, HIDP, EMBP, EMB, stream);
            run_bn(vz, vn, stat, vbn2_g + (size_t)l * EMB, vbn2_b + (size_t)l * EMB,
                   N_GRAPHS, EMBP, EMB, EMBP, 1, stream);
        } else {
            // final layer: out = BN(z), written DENSE [100000,300] into d_out
            run_bn(agg, (float*)d_out, stat, bn_g + (size_t)l * EMB, bn_b + (size_t)l * EMB,
                   N_NODES, EMBP, EMB, EMB /* dense output stride */, 0, stream);
        }
    }
}